// Extract_high_freq_by_walvet_64372969832958
// MI455X (gfx1250) — compile-verified
//
#include <hip/hip_runtime.h>

typedef float v2f __attribute__((ext_vector_type(2)));
typedef float v8f __attribute__((ext_vector_type(8)));

namespace wav {
constexpr float DL[8] = {
  -0.010597401784997278f, 0.032883011666982945f, 0.030841381835986965f,
  -0.18703481171888114f,  -0.02798376941698385f, 0.6308807679295904f,
   0.7148465705525415f,    0.23037781330885523f };
// DEC_HI[k] = (-1)^(k+1) * DEC_LO[7-k]
constexpr float DH[8] = { -DL[7], DL[6], -DL[5], DL[4], -DL[3], DL[2], -DL[1], DL[0] };
// REC_LO = reverse(DEC_LO), REC_HI = reverse(DEC_HI)
constexpr float RL[8] = { DL[7], DL[6], DL[5], DL[4], DL[3], DL[2], DL[1], DL[0] };
constexpr float RH[8] = { DH[7], DH[6], DH[5], DH[4], DH[3], DH[2], DH[1], DH[0] };
}

#define N0   512
#define MOUT 259   // (512 + 8 - 1) / 2

// pywt symmetric extension of length-512 signal: ext index (0..524) -> src index
__device__ __forceinline__ int mirror_ext(int c) {
  return (c < 6) ? (5 - c) : ((c < 518) ? (c - 6) : (1029 - c));
}

// DWT band weight: coefficient for tap offset t (= ext_pos - 2*j), 0 outside [0,8)
__device__ __forceinline__ float dwt_tap(const float* w, int t) {
  float r = 0.0f;
#pragma unroll
  for (int i = 0; i < 8; ++i) r = (t == i) ? w[7 - i] : r;
  return r;
}

// IDWT band weight: out[x] = sum_{p=0..3} a[(x>>1)+p] * c[p]; c depends on parity of x
__device__ __forceinline__ float idwt_tap(const float* w, int par, int t) {
  float r = 0.0f;
#pragma unroll
  for (int i = 0; i < 4; ++i) {
    float c = par ? w[7 - 2 * i] : w[6 - 2 * i];
    r = (t == i) ? c : r;
  }
  return r;
}

// One 16(j) x 16(row) DWT tile: 10 B fetches + 20 WMMAs.
// TAIL=false: cb in-bounds by construction -> pure base+immediate LDS addressing.
// TAIL=true (jb==16): clamp reads; polluted output rows are discarded at store.
template<bool TAIL>
__device__ __forceinline__ void dwt_tile(
    const float (&tile)[16][540], int jj, int half, int e0,
    const v2f (&aL)[10], const v2f (&aH)[10], v8f& accL, v8f& accH) {
  v2f b[10];
#pragma unroll
  for (int q = 0; q < 10; ++q) {
    int cb = e0 + 4 * q + 2 * half;
    int c0 = TAIL ? ((cb     < 524) ? cb     : 524) : cb;
    int c1 = TAIL ? ((cb + 1 < 524) ? cb + 1 : 524) : (cb + 1);
    b[q].x = tile[jj][2 + c0];
    b[q].y = tile[jj][2 + c1];
  }
#pragma unroll
  for (int q = 0; q < 10; ++q) {
    accL = __builtin_amdgcn_wmma_f32_16x16x4_f32(false, aL[q], false, b[q],
                                                 (short)0, accL, false, false);
    accH = __builtin_amdgcn_wmma_f32_16x16x4_f32(false, aH[q], false, b[q],
                                                 (short)0, accH, false, false);
  }
}

template<bool GUARD>
__device__ __forceinline__ void dwt_store(
    float* __restrict__ pa, float* __restrict__ pd,
    const v8f& accL, const v8f& accH, int j0, int half) {
#pragma unroll
  for (int v = 0; v < 8; ++v) {
    int j = j0 + 8 * half + v;        // D layout: M = vgpr + 8*half, N = lane&15
    if (!GUARD || j < MOUT) { pa[j] = accL[v]; pd[j] = accH[v]; }
  }
}

// ---------------------------------------------------------------------------
// Stage 1: row DWT via V_WMMA_F32_16X16X4_F32.
// Block = 256 threads (8 waves), 16 rows of one image. The symmetric extension
// (525 cols) is materialized in LDS once, so the WMMA loop is branch-free.
// LDS row stride 540 floats: 540 % 64 == 28, gcd(28,64)=4 -> 16 rows hit 16
// distinct banks; the +2-column lane group lands on the disjoint (mod 4) set.
// Banded A-matrix coefficients come from a small zero-padded LDS table: each
// lane's fragment is one ds_load_b64 instead of an 8-way cndmask chain.
// ---------------------------------------------------------------------------
__global__ __launch_bounds__(256) void row_dwt_wmma(
    const float* __restrict__ x, float* __restrict__ outA, float* __restrict__ outD) {
  __shared__ float tile[16][540];   // ext value e stored at column e + 2
  __shared__ float tabL[72];        // tabL[32 + t] = dwt_tap(DL, t), t in [-32,40)
  __shared__ float tabH[72];
  const int rb  = blockIdx.x;       // 0..31  (row block)
  const int img = blockIdx.y;
  const int tid = threadIdx.x;
  const int r0  = rb * 16;
  const float* src = x + img * (N0 * N0) + r0 * N0;

  // bulk copy: x[i] -> tile[row][8 + i]  (ext index i+6).  (8+4k)*4 bytes: 8B aligned
  for (int i = tid; i < 16 * 128; i += 256) {
    int row = i >> 7, c4 = i & 127;
    float4 v = *reinterpret_cast<const float4*>(src + row * N0 + c4 * 4);
    *reinterpret_cast<float2*>(&tile[row][8 + c4 * 4])     = make_float2(v.x, v.y);
    *reinterpret_cast<float2*>(&tile[row][8 + c4 * 4 + 2]) = make_float2(v.z, v.w);
  }
  // halo: ext cols 0..5 and 518..524 (13 per row, 208 total)
  if (tid < 16 * 13) {
    int row = tid / 13, h = tid - row * 13;
    int ce  = (h < 6) ? h : (512 + h);            // 518..524 for h=6..12
    tile[row][ce + 2] = src[row * N0 + mirror_ext(ce)];
  }
  // coefficient tables (once per block)
  if (tid >= 208 && tid < 208 + 72) {
    int t = (tid - 208) - 32;
    tabL[tid - 208] = dwt_tap(wav::DL, t);
    tabH[tid - 208] = dwt_tap(wav::DH, t);
  }
  __syncthreads();

  const int lane = tid & 31;
  const int wave = tid >> 5;
  const int jj   = lane & 15;       // A-matrix M row == B-matrix N column index
  const int half = lane >> 4;       // selects K pair {0,1} vs {2,3}

  // A fragments: tb even -> 8B-aligned ds_load_b64 per fragment.
  const int tb = 32 + 2 * half - 2 * jj;     // in [2, 34]
  v2f aL[10], aH[10];
#pragma unroll
  for (int q = 0; q < 10; ++q) {
    aL[q] = *reinterpret_cast<const v2f*>(&tabL[tb + 4 * q]);
    aH[q] = *reinterpret_cast<const v2f*>(&tabH[tb + 4 * q]);
  }

  const int outBase = img * (N0 * MOUT);
  float* pa = outA + outBase + (r0 + jj) * MOUT;
  float* pd = outD + outBase + (r0 + jj) * MOUT;

  // Full tiles: every wave runs jb = wave and wave+8 (both < 16): no clamps,
  // no store guards -> LDS reads become base + immediate-offset.
#pragma unroll
  for (int t = 0; t < 2; ++t) {
    const int jb = wave + 8 * t;
    v8f accL = {}; v8f accH = {};
    dwt_tile<false>(tile, jj, half, 32 * jb, aL, aH, accL, accH);
    dwt_store<false>(pa, pd, accL, accH, 16 * jb, half);
  }
  // Tail tile (j = 256..258): wave 0 only, clamped reads + guarded stores.
  if (wave == 0) {
    v8f accL = {}; v8f accH = {};
    dwt_tile<true>(tile, jj, half, 32 * 16, aL, aH, accL, accH);
    dwt_store<true>(pa, pd, accL, accH, 16 * 16, half);
  }
}

// ---------------------------------------------------------------------------
// Stage 2: column DWT (lane = column -> fully coalesced). One input -> a,d.
// ---------------------------------------------------------------------------
__global__ __launch_bounds__(64) void col_dwt(
    const float* __restrict__ in, float* __restrict__ outA, float* __restrict__ outD) {
  int c = blockIdx.x * 64 + threadIdx.x;
  int i = blockIdx.y, img = blockIdx.z;
  if (c >= MOUT) return;
  const float* src = in + img * (N0 * MOUT);
  float a = 0.f, d = 0.f;
#pragma unroll
  for (int k = 0; k < 8; ++k) {
    int r = mirror_ext(2 * i + k);
    float v = src[r * MOUT + c];
    a = fmaf(v, wav::DL[7 - k], a);
    d = fmaf(v, wav::DH[7 - k], d);
  }
  int o = img * (MOUT * MOUT) + i * MOUT + c;
  outA[o] = a; outD[o] = d;
}

// ---------------------------------------------------------------------------
// Stage 3: column IDWT: out[y] = sum_p A[(y>>1)+p]*sA*cA[p] + D[..]*sD*cD[p]
// ---------------------------------------------------------------------------
__global__ __launch_bounds__(64) void col_idwt(
    const float* __restrict__ inA, const float* __restrict__ inD,
    float* __restrict__ out, float sA, float sD) {
  int c = blockIdx.x * 64 + threadIdx.x;
  int y = blockIdx.y, img = blockIdx.z;
  if (c >= MOUT) return;
  const float* A = inA + img * (MOUT * MOUT);
  const float* D = inD + img * (MOUT * MOUT);
  int base = y >> 1, par = y & 1;
  float acc = 0.f;
#pragma unroll
  for (int p = 0; p < 4; ++p) {
    float ca = par ? wav::RL[7 - 2 * p] : wav::RL[6 - 2 * p];
    float cd = par ? wav::RH[7 - 2 * p] : wav::RH[6 - 2 * p];
    int o = (base + p) * MOUT + c;
    acc = fmaf(A[o], sA * ca, acc);
    acc = fmaf(D[o], sD * cd, acc);
  }
  out[img * (N0 * MOUT) + y * MOUT + c] = acc;
}

// ---------------------------------------------------------------------------
// Stage 4: row IDWT via WMMA. 6 WMMA per 16(x) x 16(row) output tile.
// Taps: out[x] = sum_{p<4} lo[(x>>1)+p]*cA[p] + hi[(x>>1)+p]*cD[p]
// LDS stride 260: 260 % 64 == 4 -> conflict-free as in stage 1.
// Parity-dependent A coefficients from a 2x24-entry LDS table per filter.
// ---------------------------------------------------------------------------
__global__ __launch_bounds__(256) void row_idwt_wmma(
    const float* __restrict__ lo, const float* __restrict__ hi,
    float* __restrict__ out) {
  __shared__ float tlo[16][260];
  __shared__ float thi[16][260];
  __shared__ float tabRL[48];       // [par*24 + 8 + t] = idwt_tap(RL, par, t)
  __shared__ float tabRH[48];
  const int rb = blockIdx.x, img = blockIdx.y, tid = threadIdx.x;
  const int r0 = rb * 16;
  const int inBase = img * (N0 * MOUT) + r0 * MOUT;
  for (int i = tid; i < 16 * 260; i += 256) {
    int row = i / 260, c = i - row * 260;
    float vl = 0.f, vh = 0.f;                 // pad col 259 zeroed (coeff 0 anyway)
    if (c < MOUT) { vl = lo[inBase + row * MOUT + c]; vh = hi[inBase + row * MOUT + c]; }
    tlo[row][c] = vl; thi[row][c] = vh;
  }
  if (tid < 48) {
    int p = tid / 24, t = (tid % 24) - 8;
    tabRL[tid] = idwt_tap(wav::RL, p, t);
    tabRH[tid] = idwt_tap(wav::RH, p, t);
  }
  __syncthreads();

  const int lane = tid & 31, wave = tid >> 5;
  const int xq = lane & 15, half = lane >> 4;
  const int par = xq & 1, lb = xq >> 1;

  const int tb = par * 24 + 8 + 2 * half - lb;   // +4q+1 stays in [1, 19]
  v2f aL[3], aH[3];
#pragma unroll
  for (int q = 0; q < 3; ++q) {
    aL[q].x = tabRL[tb + 4 * q]; aL[q].y = tabRL[tb + 4 * q + 1];
    aH[q].x = tabRH[tb + 4 * q]; aH[q].y = tabRH[tb + 4 * q + 1];
  }

#pragma unroll
  for (int t = 0; t < 4; ++t) {               // 32 x-blocks / 8 waves = 4 each
    const int xb = wave + 8 * t;
    const int b0 = 8 * xb;                    // input base = x0 >> 1
    v2f bl[3], bh[3];
#pragma unroll
    for (int q = 0; q < 3; ++q) {
      int cc = b0 + 4 * q + 2 * half;         // max 259 -> inside padded row
      bl[q].x = tlo[xq][cc]; bl[q].y = tlo[xq][cc + 1];
      bh[q].x = thi[xq][cc]; bh[q].y = thi[xq][cc + 1];
    }
    v8f acc = {};
#pragma unroll
    for (int q = 0; q < 3; ++q) {
      acc = __builtin_amdgcn_wmma_f32_16x16x4_f32(false, aL[q], false, bl[q],
                                                  (short)0, acc, false, false);
      acc = __builtin_amdgcn_wmma_f32_16x16x4_f32(false, aH[q], false, bh[q],
                                                  (short)0, acc, false, false);
    }
    float* po = out + img * (N0 * N0) + (r0 + xq) * N0 + 16 * xb + 8 * half;
    *reinterpret_cast<float4*>(po)     = make_float4(acc[0], acc[1], acc[2], acc[3]);
    *reinterpret_cast<float4*>(po + 4) = make_float4(acc[4], acc[5], acc[6], acc[7]);
  }
}

// ---------------------------------------------------------------------------
extern "C" void kernel_launch(void* const* d_in, const int* in_sizes, int n_in,
                              void* d_out, int out_size, void* d_ws, size_t ws_size,
                              hipStream_t stream) {
  (void)n_in; (void)out_size; (void)ws_size;
  const float* x = (const float*)d_in[0];
  float* out = (float*)d_out;
  float* ws  = (float*)d_ws;

  const int nimg = in_sizes[0] / (N0 * N0);            // 96
  const size_t S1 = (size_t)nimg * N0 * MOUT;          // row-DWT band size
  const size_t S2 = (size_t)nimg * MOUT * MOUT;        // subband size

  float* lo_w = ws;
  float* hi_w = ws + S1;
  float* ll = ws + 2 * S1;
  float* lh = ll + S2;
  float* hl = lh + S2;
  float* hh = hl + S2;
  float* lo_rec = lo_w;   // reuse: lo_w/hi_w dead after stage 2
  float* hi_rec = hi_w;

  dim3 b1(256), g1(32, nimg);
  row_dwt_wmma<<<g1, b1, 0, stream>>>(x, lo_w, hi_w);

  dim3 b2(64);
  dim3 g2((MOUT + 63) / 64, MOUT, nimg);
  col_dwt<<<g2, b2, 0, stream>>>(lo_w, ll, lh);
  col_dwt<<<g2, b2, 0, stream>>>(hi_w, hl, hh);

  dim3 g3((MOUT + 63) / 64, N0, nimg);
  col_idwt<<<g3, b2, 0, stream>>>(ll, lh, lo_rec, 1.0f, 2.0f);
  col_idwt<<<g3, b2, 0, stream>>>(hl, hh, hi_rec, 2.0f, 2.0f);

  row_idwt_wmma<<<g1, b1, 0, stream>>>(lo_rec, hi_rec, out);
}